// Fattention_88862873354867
// MI455X (gfx1250) — compile-verified
//
#include <hip/hip_runtime.h>
#include <hip/hip_bf16.h>
#include <math.h>

typedef __attribute__((ext_vector_type(2))) float v2f;
typedef __attribute__((ext_vector_type(8))) float v8f;

#define NB 8
#define NC 64
#define NF 32
#define HW 4096            // 64*64
#define CHW (NC * HW)      // 262144 elements per (b,f)
#define NEG_BIG (-3.0e38f)

// ---------- top-4 insertion (t0 >= t1 >= t2 >= t3) ----------
__device__ __forceinline__ void ins4(float v, float& t0, float& t1, float& t2, float& t3) {
    if (v > t3) {
        if (v > t1) {
            if (v > t0) { t3 = t2; t2 = t1; t1 = t0; t0 = v; }
            else        { t3 = t2; t2 = t1; t1 = v; }
        } else {
            if (v > t2) { t3 = t2; t2 = v; }
            else        { t3 = v; }
        }
    }
}

// ============================================================
// Kernel 1: per-(b,f) reduction -> topk_sum (rows 0..7) and avg (rows 8..15)
// of the stacked 16x32 matrix M stored row-major in ws[0..511].
// One block per (b,f); 256 threads stream 64 c-slices of 4096 contiguous f32.
// ============================================================
__global__ void __launch_bounds__(256)
reduce_kernel(const float* __restrict__ x, float* __restrict__ Mmat) {
    const int bf = blockIdx.x;
    const int b  = bf >> 5;
    const int f  = bf & 31;
    const int t  = threadIdx.x;

    // x[b][c][f][hw] : base of c=0 slice for this (b,f)
    const float* base = x + ((size_t)(b * NC) * NF + f) * (size_t)HW;

    float sum = 0.0f;
    float t0 = NEG_BIG, t1 = NEG_BIG, t2 = NEG_BIG, t3 = NEG_BIG;

    #pragma unroll 4
    for (int c = 0; c < NC; ++c) {
        const float4* p = (const float4*)(base + (size_t)c * (NF * HW));
        // 1024 float4 per slice, 256 threads -> 4 float4 each, coalesced
        for (int j = t; j < HW / 4; j += 256) {
            float4 v = p[j];
            sum += (v.x + v.y) + (v.z + v.w);
            ins4(v.x, t0, t1, t2, t3);
            ins4(v.y, t0, t1, t2, t3);
            ins4(v.z, t0, t1, t2, t3);
            ins4(v.w, t0, t1, t2, t3);
        }
    }

    __shared__ float s_sum[256];
    __shared__ float s_top[256 * 4];
    __shared__ float s_top2[32 * 4];

    s_sum[t] = sum;
    s_top[t * 4 + 0] = t0;
    s_top[t * 4 + 1] = t1;
    s_top[t * 4 + 2] = t2;
    s_top[t * 4 + 3] = t3;
    __syncthreads();

    // sum tree reduction
    for (int s = 128; s > 0; s >>= 1) {
        if (t < s) s_sum[t] += s_sum[t + s];
        __syncthreads();
    }

    // stage 1: 32 lanes each merge 8 thread-lists (32 candidates) -> top4
    if (t < 32) {
        float a0 = NEG_BIG, a1 = NEG_BIG, a2 = NEG_BIG, a3 = NEG_BIG;
        for (int j = 0; j < 32; ++j) ins4(s_top[t * 32 + j], a0, a1, a2, a3);
        s_top2[t * 4 + 0] = a0;
        s_top2[t * 4 + 1] = a1;
        s_top2[t * 4 + 2] = a2;
        s_top2[t * 4 + 3] = a3;
    }
    __syncthreads();

    // stage 2: thread 0 merges 128 candidates -> final top4 sum; write M rows
    if (t == 0) {
        float a0 = NEG_BIG, a1 = NEG_BIG, a2 = NEG_BIG, a3 = NEG_BIG;
        for (int j = 0; j < 128; ++j) ins4(s_top2[j], a0, a1, a2, a3);
        float topsum = ((a0 + a1) + (a2 + a3));
        float avg = s_sum[0] * (1.0f / (float)CHW);
        Mmat[b * NF + f]        = topsum;  // rows 0..7  : topk_sum
        Mmat[(8 + b) * NF + f]  = avg;     // rows 8..15 : avg
    }
}

// ============================================================
// Kernel 2: batched MLP via V_WMMA_F32_16X16X4_F32 (one wave, 32 lanes).
//   H = relu(M @ W1 + b1)        M:16x32, W1:32x16 -> 16x16   (8 K-steps)
//   O = H @ W2 + b2              H:16x16, W2:16x32 -> 16x32   (2 tiles x 4 K-steps)
//   gate[b][f] = sigmoid(O[b][f] + O[b+8][f])
// f32 WMMA layouts (ISA 7.12.2):
//   A 16x4: lane L holds row M=L&15; (v.x,v.y) = K = (L<16 ? 0,1 : 2,3)
//   B 4x16: lane L holds col N=L&15; (v.x,v.y) = K = (L<16 ? 0,1 : 2,3)
//   C/D:    vgpr r, lane L -> D[M = r + (L>=16 ? 8:0)][N = L&15]
// ============================================================
__global__ void __launch_bounds__(32)
mlp_kernel(const float* __restrict__ W1, const float* __restrict__ b1,
           const float* __restrict__ W2, const float* __restrict__ b2,
           const float* __restrict__ Mmat,
           float* __restrict__ gate, float* __restrict__ out_gate) {
    const int lane = threadIdx.x;      // 0..31, all lanes active (EXEC all 1s)
    const int row  = lane & 15;        // A row / B col / D col
    const int hi   = lane >> 4;        // half-wave select

    __shared__ float Hs[16 * 16];
    __shared__ float Os[16 * 32];

    // ---- GEMM1: acc = M @ W1, accumulator pre-seeded with bias b1[N] ----
    v8f acc;
    {
        float bv = b1[row];
        for (int r = 0; r < 8; ++r) acc[r] = bv;
    }
    #pragma unroll
    for (int ki = 0; ki < 8; ++ki) {
        const int kb = ki * 4 + hi * 2;
        v2f a, bb;
        a.x  = Mmat[row * 32 + kb];
        a.y  = Mmat[row * 32 + kb + 1];
        bb.x = W1[kb * 16 + row];
        bb.y = W1[(kb + 1) * 16 + row];
        acc = __builtin_amdgcn_wmma_f32_16x16x4_f32(
            /*neg_a=*/false, a, /*neg_b=*/false, bb,
            /*c_mod=*/(short)0, acc, /*reuse_a=*/false, /*reuse_b=*/false);
    }

    // ReLU, spill H (D layout) to LDS for A-layout re-read
    #pragma unroll
    for (int r = 0; r < 8; ++r) {
        float h = acc[r] > 0.0f ? acc[r] : 0.0f;
        Hs[(hi * 8 + r) * 16 + row] = h;
    }
    __syncthreads();

    // ---- GEMM2: O = H @ W2 (+b2), 2 N-tiles of 16 ----
    #pragma unroll
    for (int tile = 0; tile < 2; ++tile) {
        v8f acc2;
        {
            float bv = b2[tile * 16 + row];
            for (int r = 0; r < 8; ++r) acc2[r] = bv;
        }
        #pragma unroll
        for (int ki = 0; ki < 4; ++ki) {
            const int kb = ki * 4 + hi * 2;
            v2f a, bb;
            a.x  = Hs[row * 16 + kb];
            a.y  = Hs[row * 16 + kb + 1];
            bb.x = W2[kb * 32 + tile * 16 + row];
            bb.y = W2[(kb + 1) * 32 + tile * 16 + row];
            acc2 = __builtin_amdgcn_wmma_f32_16x16x4_f32(
                false, a, false, bb, (short)0, acc2, false, false);
        }
        #pragma unroll
        for (int r = 0; r < 8; ++r)
            Os[(hi * 8 + r) * 32 + tile * 16 + row] = acc2[r];
    }
    __syncthreads();

    // ---- gate = sigmoid(O[b] + O[b+8]) : 256 values, 8 per lane ----
    for (int it = 0; it < 8; ++it) {
        const int idx = it * 32 + lane;
        const int bb_ = idx >> 5;
        const int ff  = idx & 31;
        float s = Os[bb_ * 32 + ff] + Os[(bb_ + 8) * 32 + ff];
        float g = 1.0f / (1.0f + expf(-s));
        gate[idx]     = g;   // workspace copy for scale kernel
        out_gate[idx] = g;   // second tuple output
    }
}

// ============================================================
// Kernel 3: out = x * gate[b,f]   (float4 grid-stride stream)
// flat float4 index: [b:3][c:6][f:5][hw4:10]
// ============================================================
__global__ void __launch_bounds__(256)
scale_kernel(const float* __restrict__ x, const float* __restrict__ gate,
             float* __restrict__ out) {
    const int N4 = NB * NC * NF * (HW / 4);   // 16,777,216
    const float4* __restrict__ x4 = (const float4*)x;
    float4* __restrict__ o4 = (float4*)out;
    int i = blockIdx.x * blockDim.x + threadIdx.x;
    const int stride = gridDim.x * blockDim.x;
    for (; i < N4; i += stride) {
        const int f = (i >> 10) & 31;
        const int b = i >> 21;
        const float g = gate[(b << 5) | f];
        float4 v = x4[i];
        v.x *= g; v.y *= g; v.z *= g; v.w *= g;
        o4[i] = v;
    }
}

extern "C" void kernel_launch(void* const* d_in, const int* in_sizes, int n_in,
                              void* d_out, int out_size, void* d_ws, size_t ws_size,
                              hipStream_t stream) {
    const float* x  = (const float*)d_in[0];
    const float* W1 = (const float*)d_in[1];
    const float* b1 = (const float*)d_in[2];
    const float* W2 = (const float*)d_in[3];
    const float* b2 = (const float*)d_in[4];
    float* out = (float*)d_out;
    float* ws  = (float*)d_ws;

    float* Mmat = ws;        // 16x32 stacked [topk_sum ; avg]
    float* gate = ws + 512;  // 8x32 gate

    const int XN = NB * NC * NF * HW;   // 67,108,864 elements of x

    reduce_kernel<<<NB * NF, 256, 0, stream>>>(x, Mmat);
    mlp_kernel<<<1, 32, 0, stream>>>(W1, b1, W2, b2, Mmat, gate, out + XN);
    scale_kernel<<<8192, 256, 0, stream>>>(x, gate, out);
}